// Attention_48335561949930
// MI455X (gfx1250) — compile-verified
//
#include <hip/hip_runtime.h>

typedef __attribute__((ext_vector_type(2))) float v2f;
typedef __attribute__((ext_vector_type(8))) float v8f;

#define BB 2
#define TT 512
#define DD 256
#define BTD (BB * TT * DD)   // 262144 elements
#define MM  (BB * TT)        // 1024 GEMM rows

// ---------------------------------------------------------------------------
// Phase 0: time-shift mix   xk/xv/xr = x*mix + shift(x)*(1-mix)
// ---------------------------------------------------------------------------
__global__ void rwkv_mix_kernel(const float* __restrict__ x,
                                const float* __restrict__ last_x,
                                const float* __restrict__ tmk,
                                const float* __restrict__ tmv,
                                const float* __restrict__ tmr,
                                float* __restrict__ xk,
                                float* __restrict__ xv,
                                float* __restrict__ xr) {
  int idx = blockIdx.x * blockDim.x + threadIdx.x;
  if (idx >= BTD) return;
  int d  = idx % DD;
  int bt = idx / DD;
  int t  = bt % TT;
  int b  = bt / TT;
  float xc = x[idx];
  float xp = (t == 0) ? last_x[b * DD + d] : x[idx - DD];
  float mk = tmk[d], mv = tmv[d], mr = tmr[d];
  xk[idx] = xc * mk + xp * (1.0f - mk);
  xv[idx] = xc * mv + xp * (1.0f - mv);
  xr[idx] = xc * mr + xp * (1.0f - mr);
}

// ---------------------------------------------------------------------------
// WMMA GEMM:  C[M][N] = A[M][K] @ W[N][K]^T   (M=1024, N=K=256, fp32)
// One wave32 computes a 16(M) x 64(N) strip of C with V_WMMA_F32_16X16X4_F32:
// per K-step it loads ONE A fragment and four B fragments and issues four
// WMMAs (4 wmma : 5 vmem instead of 1 : 2 with a single 16x16 tile).
//
// Fragment layouts (ISA 7.12.2, wave32):
//   A 16x4 f32 : lanes 0-15 -> M=lane, K=(k0+0,k0+1); lanes 16-31 -> M=lane-16,
//                K=(k0+2,k0+3); two consecutive VGPRs hold the two K values.
//   B 4x16 f32 : mirrored over N (lane%16 = N), same K split across lane halves.
//   C/D 16x16  : 8 VGPRs; VGPR i holds M = i + 8*(lane>=16), N = lane%16.
// ---------------------------------------------------------------------------
__global__ void wmma_gemm_awt(const float* __restrict__ A,
                              const float* __restrict__ W,
                              float* __restrict__ C) {
  const int K = DD, N = DD;
  const int lane = threadIdx.x & 31;
  const int wave = (blockIdx.x * blockDim.x + threadIdx.x) >> 5;
  const int tiles_n = N / 64;                 // 4 column-strips of 64
  const int tile_n = wave % tiles_n;
  const int tile_m = wave / tiles_n;

  const int m     = tile_m * 16 + (lane & 15);   // A row for this lane
  const int khalf = (lane >> 4) * 2;             // 0 for lanes 0-15, 2 for 16-31

  const float* __restrict__ arow = A + (size_t)m * K;
  // B columns n_j = tile_n*64 + j*16 + (lane&15); each is a row of W.
  const float* __restrict__ wrow = W + (size_t)(tile_n * 64 + (lane & 15)) * K;

  v8f acc0 = {}, acc1 = {}, acc2 = {}, acc3 = {};
  for (int k0 = 0; k0 < K; k0 += 4) {
    const int kk = k0 + khalf;
    v2f a  = *(const v2f*)(arow + kk);            // {A[m][kk], A[m][kk+1]}
    v2f b0 = *(const v2f*)(wrow + kk);            // B frag for cols j*16..
    v2f b1 = *(const v2f*)(wrow + 16 * K + kk);
    v2f b2 = *(const v2f*)(wrow + 32 * K + kk);
    v2f b3 = *(const v2f*)(wrow + 48 * K + kk);
    acc0 = __builtin_amdgcn_wmma_f32_16x16x4_f32(false, a, false, b0, (short)0, acc0, false, false);
    acc1 = __builtin_amdgcn_wmma_f32_16x16x4_f32(false, a, false, b1, (short)0, acc1, false, false);
    acc2 = __builtin_amdgcn_wmma_f32_16x16x4_f32(false, a, false, b2, (short)0, acc2, false, false);
    acc3 = __builtin_amdgcn_wmma_f32_16x16x4_f32(false, a, false, b3, (short)0, acc3, false, false);
  }

  const int row_base = tile_m * 16 + (lane >> 4) * 8;
  const int col0     = tile_n * 64 + (lane & 15);
#pragma unroll
  for (int i = 0; i < 8; ++i) {
    float* crow = C + (size_t)(row_base + i) * N + col0;
    crow[0]  = acc0[i];
    crow[16] = acc1[i];
    crow[32] = acc2[i];
    crow[48] = acc3[i];
  }
}

// ---------------------------------------------------------------------------
// Phase 2: WKV exponential scan, one thread per (b, d) channel.
// Reproduces the reference's masked-exp quadratic form via the equivalent
// linear recurrence. Emits wkv*sigmoid(r) for the final GEMM and writes the
// (x_last, num_last, den_last) output tails.
// ---------------------------------------------------------------------------
__global__ void rwkv_scan_kernel(const float* __restrict__ kbuf,
                                 const float* __restrict__ vbuf,
                                 const float* __restrict__ rbuf,
                                 const float* __restrict__ x,
                                 const float* __restrict__ last_num,
                                 const float* __restrict__ last_den,
                                 const float* __restrict__ time_decay,
                                 const float* __restrict__ time_first,
                                 float* __restrict__ wkvsr,
                                 float* __restrict__ out_tail) {
  int idx = blockIdx.x * blockDim.x + threadIdx.x;   // over B*D = 512
  if (idx >= BB * DD) return;
  int d = idx % DD;
  int b = idx / DD;

  const float w  = -__expf(time_decay[d]);
  const float ew = __expf(w);
  const float tf = time_first[d];

  float num = last_num[idx];
  float den = last_den[idx];

  const float* kp = kbuf + (size_t)b * TT * DD + d;
  const float* vp = vbuf + (size_t)b * TT * DD + d;
  const float* rp = rbuf + (size_t)b * TT * DD + d;
  float* op = wkvsr + (size_t)b * TT * DD + d;

  for (int t = 0; t < TT; ++t) {
    const size_t off = (size_t)t * DD;
    float kk = kp[off];
    float vv = vp[off];
    float rr = rp[off];
    float euk = __expf(tf + kk);
    float ek  = __expf(kk);
    float sr  = 1.0f / (1.0f + __expf(-rr));
    float wkv = (num + euk * vv) / (den + euk);
    op[off] = wkv * sr;
    num = ew * num + ek * vv;
    den = ew * den + ek;
  }

  // output tails: x[:, -1, :], num[:, -1, :], den[:, -1, :]
  out_tail[idx]               = x[(size_t)b * TT * DD + (size_t)(TT - 1) * DD + d];
  out_tail[BB * DD + idx]     = num;
  out_tail[2 * BB * DD + idx] = den;
}

// ---------------------------------------------------------------------------
// Launch
// ---------------------------------------------------------------------------
extern "C" void kernel_launch(void* const* d_in, const int* in_sizes, int n_in,
                              void* d_out, int out_size, void* d_ws, size_t ws_size,
                              hipStream_t stream) {
  const float* x         = (const float*)d_in[0];
  const float* last_x    = (const float*)d_in[1];
  const float* last_num  = (const float*)d_in[2];
  const float* last_den  = (const float*)d_in[3];
  const float* t_decay   = (const float*)d_in[4];
  const float* t_first   = (const float*)d_in[5];
  const float* tmk       = (const float*)d_in[6];
  const float* tmv       = (const float*)d_in[7];
  const float* tmr       = (const float*)d_in[8];
  const float* Wk        = (const float*)d_in[9];
  const float* Wv        = (const float*)d_in[10];
  const float* Wr        = (const float*)d_in[11];
  const float* Wo        = (const float*)d_in[12];

  float* out = (float*)d_out;

  float* ws = (float*)d_ws;
  float* xk    = ws + 0 * (size_t)BTD;
  float* xv    = ws + 1 * (size_t)BTD;
  float* xr    = ws + 2 * (size_t)BTD;
  float* kb    = ws + 3 * (size_t)BTD;
  float* vb    = ws + 4 * (size_t)BTD;
  float* rb    = ws + 5 * (size_t)BTD;
  float* wkvsr = ws + 6 * (size_t)BTD;

  // Phase 0: mixes
  rwkv_mix_kernel<<<(BTD + 255) / 256, 256, 0, stream>>>(
      x, last_x, tmk, tmv, tmr, xk, xv, xr);

  // Phase 1: k, v, r projections (WMMA fp32), 16x64 strip per wave
  const int waves  = (MM / 16) * (DD / 64);       // 256 waves
  const int blocks = (waves * 32) / 256;          // 8 waves/block -> 32 blocks
  wmma_gemm_awt<<<blocks, 256, 0, stream>>>(xk, Wk, kb);
  wmma_gemm_awt<<<blocks, 256, 0, stream>>>(xv, Wv, vb);
  wmma_gemm_awt<<<blocks, 256, 0, stream>>>(xr, Wr, rb);

  // Phase 2: WKV scan + sigmoid(r) gating + output tails
  rwkv_scan_kernel<<<(BB * DD + 255) / 256, 256, 0, stream>>>(
      kb, vb, rb, x, last_num, last_den, t_decay, t_first,
      wkvsr, out + (size_t)BTD);

  // Phase 3: output projection (WMMA fp32)
  wmma_gemm_awt<<<blocks, 256, 0, stream>>>(wkvsr, Wo, out);
}